// RhythmNetLoss_25520695673272
// MI455X (gfx1250) — compile-verified
//
#include <hip/hip_runtime.h>

// RhythmNetLoss for MI455X (gfx1250, wave32).
// Memory-bound two-pass reduction. Uses V_WMMA_F32_16X16X4_F32 (all-ones B)
// for exact-f32 cross-lane sums, NT load hints so gru survives in the 192MB L2
// for pass 2, and a partials-in-workspace scheme instead of serializing atomics.

typedef float v2f __attribute__((ext_vector_type(2)));
typedef float v4f __attribute__((ext_vector_type(4)));
typedef float v8f __attribute__((ext_vector_type(8)));

#define NB 2048   // blocks for the streaming passes
#define BS 256    // threads per block (8 waves on wave32)

// Wave-wide sum via WMMA: A = 16x4 with per-lane acc in VGPR0 (K=0 for lanes
// 0-15, K=2 for lanes 16-31), zeros in VGPR1; B = all ones (layout-agnostic).
// D[m][n] = acc[m] + acc[m+16], so sum of the 8 D VGPRs in lane l plus the
// same from lane l^16 is the full wave sum, broadcast to every lane.
__device__ __forceinline__ float wave_reduce_wmma(float v) {
    v2f a; a.x = v;    a.y = 0.0f;
    v2f b; b.x = 1.0f; b.y = 1.0f;
    v8f c = {};
    c = __builtin_amdgcn_wmma_f32_16x16x4_f32(
        /*neg_a=*/false, a, /*neg_b=*/false, b,
        /*c_mod=*/(short)0, c, /*reuse_a=*/false, /*reuse_b=*/false);
    float s = ((c[0] + c[1]) + (c[2] + c[3])) + ((c[4] + c[5]) + (c[6] + c[7]));
    s += __shfl_xor(s, 16, 32);
    return s;
}

// Returns full block sum in thread 0 (other threads: 0).
__device__ __forceinline__ float block_reduce(float v, float* lds) {
    float w = wave_reduce_wmma(v);
    const int lane = threadIdx.x & 31;
    const int wid  = threadIdx.x >> 5;
    if (lane == 0) lds[wid] = w;
    __syncthreads();
    float t = 0.0f;
    if (threadIdx.x == 0) {
        const int nw = (int)(blockDim.x >> 5);
        for (int i = 0; i < nw; ++i) t += lds[i];
    }
    return t;
}

// Pass 1: per-block partials of sum|r-t| and sum(g).
// r,t loaded non-temporal (streamed once, keep out of L2); g regular so the
// 64MB gru array parks in the 192MB L2 for pass 2.
__global__ void pass1_kernel(const float* __restrict__ r, const float* __restrict__ g,
                             const float* __restrict__ t, float* __restrict__ pL1,
                             float* __restrict__ pG, int n) {
    __shared__ float ldsA[BS / 32];
    __shared__ float ldsB[BS / 32];
    const int n4 = n >> 2;
    const v4f* r4 = (const v4f*)r;
    const v4f* g4 = (const v4f*)g;
    const v4f* t4 = (const v4f*)t;
    float accL = 0.0f, accG = 0.0f;
    for (int i = blockIdx.x * blockDim.x + threadIdx.x; i < n4;
         i += gridDim.x * blockDim.x) {
        v4f rv = __builtin_nontemporal_load(&r4[i]);
        v4f tv = __builtin_nontemporal_load(&t4[i]);
        v4f gv = g4[i];
        accL += fabsf(rv.x - tv.x) + fabsf(rv.y - tv.y) +
                fabsf(rv.z - tv.z) + fabsf(rv.w - tv.w);
        accG += (gv.x + gv.y) + (gv.z + gv.w);
    }
    if (blockIdx.x == 0 && threadIdx.x == 0) {  // scalar tail (n % 4)
        for (int i = n4 << 2; i < n; ++i) { accL += fabsf(r[i] - t[i]); accG += g[i]; }
    }
    float bL = block_reduce(accL, ldsA);
    float bG = block_reduce(accG, ldsB);
    if (threadIdx.x == 0) { pL1[blockIdx.x] = bL; pG[blockIdx.x] = bG; }
}

// Reduce pass-1 partials: scal[0] = sum|r-t|, scal[1] = mean(g).
__global__ void reduce1_kernel(const float* __restrict__ pL1, const float* __restrict__ pG,
                               float* __restrict__ scal, int nb, float invN) {
    __shared__ float ldsA[BS / 32];
    __shared__ float ldsB[BS / 32];
    float aL = 0.0f, aG = 0.0f;
    for (int i = threadIdx.x; i < nb; i += blockDim.x) { aL += pL1[i]; aG += pG[i]; }
    float sL = block_reduce(aL, ldsA);
    float sG = block_reduce(aG, ldsB);
    if (threadIdx.x == 0) { scal[0] = sL; scal[1] = sG * invN; }
}

// Pass 2: per-block partials of sum|g - mean|; g reads should hit L2.
__global__ void pass2_kernel(const float* __restrict__ g, const float* __restrict__ scal,
                             float* __restrict__ pDev, int n) {
    __shared__ float ldsA[BS / 32];
    const float mean = scal[1];
    const int n4 = n >> 2;
    const v4f* g4 = (const v4f*)g;
    float acc = 0.0f;
    for (int i = blockIdx.x * blockDim.x + threadIdx.x; i < n4;
         i += gridDim.x * blockDim.x) {
        v4f gv = g4[i];
        acc += fabsf(gv.x - mean) + fabsf(gv.y - mean) +
               fabsf(gv.z - mean) + fabsf(gv.w - mean);
    }
    if (blockIdx.x == 0 && threadIdx.x == 0) {
        for (int i = n4 << 2; i < n; ++i) acc += fabsf(g[i] - mean);
    }
    float b = block_reduce(acc, ldsA);
    if (threadIdx.x == 0) pDev[blockIdx.x] = b;
}

// Final: out = sum|r-t|/N + 100 * (sum|g-mean|/N).
__global__ void finalize_kernel(const float* __restrict__ pDev, const float* __restrict__ scal,
                                float* __restrict__ out, int nb, float invN) {
    __shared__ float ldsA[BS / 32];
    float a = 0.0f;
    for (int i = threadIdx.x; i < nb; i += blockDim.x) a += pDev[i];
    float sDev = block_reduce(a, ldsA);
    if (threadIdx.x == 0) out[0] = scal[0] * invN + 100.0f * (sDev * invN);
}

extern "C" void kernel_launch(void* const* d_in, const int* in_sizes, int n_in,
                              void* d_out, int out_size, void* d_ws, size_t ws_size,
                              hipStream_t stream) {
    const float* r = (const float*)d_in[0];   // resnet_outputs
    const float* g = (const float*)d_in[1];   // gru_outputs
    const float* t = (const float*)d_in[2];   // target
    float* out = (float*)d_out;
    float* ws  = (float*)d_ws;

    const int n = in_sizes[0];
    const float invN = 1.0f / (float)n;

    // Workspace layout (floats): [0,NB) l1 partials, [NB,2NB) g partials,
    // [2NB,3NB) dev partials, [3NB..] scalars {sum_l1, mean_g}. ~24.6 KB.
    float* pL1  = ws;
    float* pG   = ws + NB;
    float* pDev = ws + 2 * NB;
    float* scal = ws + 3 * NB;

    pass1_kernel   <<<NB, BS, 0, stream>>>(r, g, t, pL1, pG, n);
    reduce1_kernel <<<1,  BS, 0, stream>>>(pL1, pG, scal, NB, invN);
    pass2_kernel   <<<NB, BS, 0, stream>>>(g, scal, pDev, n);
    finalize_kernel<<<1,  BS, 0, stream>>>(pDev, scal, out, NB, invN);
}